// LSTMModel_82343112999199
// MI455X (gfx1250) — compile-verified
//
#include <hip/hip_runtime.h>
#include <hip/hip_bf16.h>
#include <math.h>
#include <stdint.h>

typedef __attribute__((ext_vector_type(2))) float v2f;
typedef __attribute__((ext_vector_type(8))) float v8f;

#define HID 2048
#define TILE_M 16
#define WAVES_PER_WG 4
#define WG_THREADS (WAVES_PER_WG * 32)

// ---------------------------------------------------------------------------
// GEMV over the needed LSTM gate rows (i, g, o -- the f gate is dead because
// c0 == 0). One workgroup owns a 16-row tile; K is split across 4 waves; each
// wave accumulates with V_WMMA_F32_16X16X4_F32 (full fp32 matrix pipe).
//   A (16x4, MxK)  : W tile.  lane m   (0..15): v0=W[r0+m,k+0], v1=W[r0+m,k+1]
//                             lane m+16       : v0=W[r0+m,k+2], v1=W[r0+m,k+3]
//   B (4x16, KxN)  : x[k..k+3] replicated over all 16 columns (LDS broadcast)
//   D (16x16)      : every column = partial y; y[r0+v] in vgpr v, lanes 0..15,
//                    y[r0+8+v] in vgpr v, lanes 16..31.
// x is staged into LDS with CDNA5 async-to-LDS DMA (ASYNCcnt-tracked);
// the streaming weight loads use non-temporal hints (single-pass, 167 MB).
// ---------------------------------------------------------------------------
__global__ void gemv_gates_wmma(const float* __restrict__ W,
                                const float* __restrict__ x,
                                const float* __restrict__ b_ih,
                                const float* __restrict__ b_hh,
                                float* __restrict__ y,
                                int K)
{
    __shared__ __attribute__((aligned(16))) float xs[HID];  // staged x (<=8KB)
    __shared__ float partial[WAVES_PER_WG][TILE_M];         // cross-wave reduce

    const int tid  = threadIdx.x;        // 0..127
    const int wave = tid >> 5;           // 0..3 (wave32)
    const int lane = tid & 31;

    // --- Stage x into LDS via async DMA (global_load_async_to_lds_b128). ---
    // VDST = LDS byte offset (HW adds LDS_BASE), VADDR = 64-bit global addr.
    // K is a multiple of 512 -> uniform trip count, no divergence.
    for (int i = tid * 4; i < K; i += WG_THREADS * 4) {
        const float* gp = x + i;
        unsigned lds_off = (unsigned)(uintptr_t)(&xs[i]);
        asm volatile("global_load_async_to_lds_b128 %0, %1, off"
                     :: "v"(lds_off), "v"(gp)
                     : "memory");
    }
    asm volatile("s_wait_asynccnt 0x0" ::: "memory");
    __syncthreads();

    // Tile -> gate-row mapping that skips the dead f gate.
    // gate 0 -> W rows [0,H)   (i), gate 1 -> [2H,3H) (g), gate 2 -> [3H,4H) (o)
    const int tilesPerGate = HID / TILE_M;               // 128
    const int t         = blockIdx.x;                    // 0..383
    const int gate      = t / tilesPerGate;              // 0,1,2
    const int rowInGate = (t - gate * tilesPerGate) * TILE_M;
    const int gateRowOff = (gate == 0) ? 0 : ((gate == 1) ? 2 * HID : 3 * HID);
    const int r0 = gateRowOff + rowInGate;               // row base in W

    // This wave's K range.
    const int kPerWave = K / WAVES_PER_WG;
    const int k0 = wave * kPerWave;

    // Per-lane A/B addressing per the 16x4 f32 WMMA layout.
    const int m     = lane & 15;
    const int khalf = (lane >> 4) << 1;                  // 0 or 2
    const float* wp = W + (size_t)(r0 + m) * (size_t)K + (size_t)(k0 + khalf);
    const float* xp = xs + k0 + khalf;

    v8f acc0 = (v8f)0.0f;
    v8f acc1 = (v8f)0.0f;

    // Two independent accumulator chains to pipeline WMMA latency.
    // Weight loads are non-temporal: read-once 167MB stream, keep L2 for the
    // reused small buffers.
    for (int k = 0; k < kPerWave; k += 8) {
        v2f a0 = __builtin_nontemporal_load((const v2f*)(wp + k));
        v2f b0 = *(const v2f*)(xp + k);
        v2f a1 = __builtin_nontemporal_load((const v2f*)(wp + k + 4));
        v2f b1 = *(const v2f*)(xp + k + 4);
        acc0 = __builtin_amdgcn_wmma_f32_16x16x4_f32(false, a0, false, b0,
                                                     (short)0, acc0, false, false);
        acc1 = __builtin_amdgcn_wmma_f32_16x16x4_f32(false, a1, false, b1,
                                                     (short)0, acc1, false, false);
    }
    v8f acc = acc0 + acc1;

    // Column 0 of each half-wave carries the 16 partial sums.
    if (lane == 0) {
        #pragma unroll
        for (int v = 0; v < 8; ++v) partial[wave][v] = acc[v];
    }
    if (lane == 16) {
        #pragma unroll
        for (int v = 0; v < 8; ++v) partial[wave][8 + v] = acc[v];
    }
    __syncthreads();

    // Reduce across waves, fuse both biases, write compact gate output:
    // y[0:H)=i_pre, y[H:2H)=g_pre, y[2H:3H)=o_pre.
    if (tid < TILE_M) {
        float s = partial[0][tid] + partial[1][tid]
                + partial[2][tid] + partial[3][tid];
        const int r = r0 + tid;
        s += b_ih[r] + b_hh[r];
        y[t * TILE_M + tid] = s;
    }
}

// ---------------------------------------------------------------------------
// Fused LSTM activation with c0 = 0:  h = sigmoid(o) * tanh( sigmoid(i)*tanh(g) )
// ---------------------------------------------------------------------------
__global__ void lstm_act(const float* __restrict__ y, float* __restrict__ h)
{
    const int j = blockIdx.x * blockDim.x + threadIdx.x;
    if (j >= HID) return;
    const float ig = y[j];
    const float gg = y[HID + j];
    const float og = y[2 * HID + j];
    const float i  = 1.0f / (1.0f + expf(-ig));
    const float o  = 1.0f / (1.0f + expf(-og));
    const float g  = tanhf(gg);
    const float c  = i * g;
    h[j] = o * tanhf(c);
}

// ---------------------------------------------------------------------------
// Final FC: out[0] = dot(fc_w, h) + fc_b   (2048-wide, one workgroup)
// ---------------------------------------------------------------------------
__global__ void fc_dot(const float* __restrict__ h,
                       const float* __restrict__ w,
                       const float* __restrict__ b,
                       float* __restrict__ out)
{
    __shared__ float red[256];
    const int tid = threadIdx.x;
    float s = 0.0f;
    for (int j = tid; j < HID; j += 256) s += h[j] * w[j];
    red[tid] = s;
    __syncthreads();
    for (int st = 128; st > 0; st >>= 1) {
        if (tid < st) red[tid] += red[tid + st];
        __syncthreads();
    }
    if (tid == 0) out[0] = red[0] + b[0];
}

// ---------------------------------------------------------------------------
// Input order (setup_inputs dict):
// 0:x 1:W_ih_0 2:W_hh_0 3:b_ih_0 4:b_hh_0 5:W_ih_1 6:W_hh_1 7:b_ih_1 8:b_hh_1
// 9:W_ih_2 10:W_hh_2 11:b_ih_2 12:b_hh_2 13:fc_w 14:fc_b
// W_hh_* are mathematically unused (h0 == 0).
// ---------------------------------------------------------------------------
extern "C" void kernel_launch(void* const* d_in, const int* in_sizes, int n_in,
                              void* d_out, int out_size, void* d_ws, size_t ws_size,
                              hipStream_t stream)
{
    (void)in_sizes; (void)n_in; (void)out_size; (void)ws_size;

    const float* x    = (const float*)d_in[0];
    const float* Wih0 = (const float*)d_in[1];
    const float* bih0 = (const float*)d_in[3];
    const float* bhh0 = (const float*)d_in[4];
    const float* Wih1 = (const float*)d_in[5];
    const float* bih1 = (const float*)d_in[7];
    const float* bhh1 = (const float*)d_in[8];
    const float* Wih2 = (const float*)d_in[9];
    const float* bih2 = (const float*)d_in[11];
    const float* bhh2 = (const float*)d_in[12];
    const float* fcw  = (const float*)d_in[13];
    const float* fcb  = (const float*)d_in[14];

    float* ws = (float*)d_ws;
    float* y  = ws;              // 3*HID floats
    float* ha = ws + 3 * HID;    // HID floats
    float* hb = ws + 4 * HID;    // HID floats
    float* out = (float*)d_out;

    const dim3 blk(WG_THREADS);
    const dim3 grid(3 * HID / TILE_M);   // 384 tiles (i,g,o gates)

    // Layer 0 (K = 1024, x input)
    gemv_gates_wmma<<<grid, blk, 0, stream>>>(Wih0, x, bih0, bhh0, y, 1024);
    lstm_act<<<HID / 256, 256, 0, stream>>>(y, ha);
    // Layer 1 (K = 2048)
    gemv_gates_wmma<<<grid, blk, 0, stream>>>(Wih1, ha, bih1, bhh1, y, 2048);
    lstm_act<<<HID / 256, 256, 0, stream>>>(y, hb);
    // Layer 2 (K = 2048)
    gemv_gates_wmma<<<grid, blk, 0, stream>>>(Wih2, hb, bih2, bhh2, y, 2048);
    lstm_act<<<HID / 256, 256, 0, stream>>>(y, ha);
    // Head
    fc_dot<<<1, 256, 0, stream>>>(ha, fcw, fcb, out);
}